// FFTLowFreqSelector_56564719288954
// MI455X (gfx1250) — compile-verified
//
#include <hip/hip_runtime.h>
#include <math.h>
#include <vector>
#include <algorithm>

typedef float v2f __attribute__((ext_vector_type(2)));
typedef float v8f __attribute__((ext_vector_type(8)));

#define NBINS 25
#define IMG_STRIDE 132   // 128 + 4 pad: rows stay 16B-aligned for async b128, banks conflict-free
#define R_STRIDE 17      // odd stride -> stage-2 reduction loads conflict-free

#if defined(__gfx1250__) && __has_builtin(__builtin_amdgcn_global_load_async_to_lds_b128)
#define HAVE_ASYNC_LDS 1
#else
#define HAVE_ASYNC_LDS 0
#endif

#if HAVE_ASYNC_LDS
// Builtin signature (from hipcc diagnostic): (v4i addrspace(1)*, v4i addrspace(3)*, imm, imm)
typedef int v4i_vec __attribute__((vector_size(16)));
typedef __attribute__((address_space(1))) v4i_vec* as1_v4i_p;
typedef __attribute__((address_space(3))) v4i_vec* as3_v4i_p;
#endif

struct FParams {
    int kxv[8];          // distinct signed kx frequencies (|k| <= 64)
    int kyv[8];          // distinct signed ky frequencies
    int binq[NBINS];     // per-bin index into kxv
    int binp[NBINS];     // per-bin index into kyv
    int Q, P;            // number of valid entries in kxv / kyv
};

// Issue the (async) copy of one 64KB channel into an LDS image buffer.
__device__ __forceinline__ void issue_channel_load(const float* __restrict__ gbase,
                                                   float* __restrict__ dstimg, int t) {
#pragma unroll
    for (int it = 0; it < 16; ++it) {
        int idx = it * 256 + t;                 // float4 index within channel
        int row = idx >> 5;                     // (idx*4)/128
        int col = (idx & 31) << 2;
#if HAVE_ASYNC_LDS
        __builtin_amdgcn_global_load_async_to_lds_b128(
            (as1_v4i_p)(gbase + (size_t)idx * 4),
            (as3_v4i_p)&dstimg[row * IMG_STRIDE + col],
            0, 0);
#else
        float4 v = ((const float4*)gbase)[idx];
        float* d = &dstimg[row * IMG_STRIDE + col];
        d[0] = v.x; d[1] = v.y; d[2] = v.z; d[3] = v.w;
#endif
    }
}

__device__ __forceinline__ void wait_async_done() {
#if HAVE_ASYNC_LDS
#if __has_builtin(__builtin_amdgcn_s_wait_asynccnt)
    __builtin_amdgcn_s_wait_asynccnt(0);
#else
    asm volatile("s_wait_asynccnt 0" ::: "memory");
#endif
#endif
}

// One workgroup per batch image. 256 threads = 8 waves (wave32).
// LDS layout (floats): img0[128][132] | img1[128][132] | Fx[128][16] | Gy[128][16]
//                      | R[128][17] | part[32][8]
__global__ __launch_bounds__(256) void fftlow_kernel(const float* __restrict__ x,
                                                     float* __restrict__ out,
                                                     FParams prm) {
    const int t = threadIdx.x;
    const int b = blockIdx.x;

    extern __shared__ float sm[];
    float* img0 = sm;                               // 16896
    float* img1 = img0 + 128 * IMG_STRIDE;          // 16896
    float* Fx   = img1 + 128 * IMG_STRIDE;          // 2048: Fx[x*16+2q]=cos, +1=sin
    float* Gy   = Fx + 2048;                        // 2048
    float* Rm   = Gy + 2048;                        // 128*17 = 2176
    float* part = Rm + 128 * R_STRIDE;              // 256

    const float PI2_OVER_N = 6.283185307179586f / 128.0f;
    const float* xbase = x + (size_t)b * 3 * 16384;

    // Kick off channel 0 DMA before doing any ALU setup work.
    issue_channel_load(xbase, img0, t);

    // Build twiddle tables (channel-independent) while channel 0 streams in.
    for (int i = t; i < 2048; i += 256) {
        int row = i >> 4, j = i & 15, q = j >> 1;
        float fv = 0.0f, gv = 0.0f;
        if (q < prm.Q) {
            float th = -PI2_OVER_N * (float)(prm.kxv[q] * row);
            fv = (j & 1) ? sinf(th) : cosf(th);
        }
        if (q < prm.P) {
            float th = -PI2_OVER_N * (float)(prm.kyv[q] * row);
            gv = (j & 1) ? sinf(th) : cosf(th);
        }
        Fx[i] = fv;
        Gy[i] = gv;
    }

    const int lane = t & 31;
    const int w    = t >> 5;        // wave id -> M tile
    const int lg   = lane & 15;     // lane within half
    const int hi   = lane >> 4;     // lane-half select
    float acc = 0.0f;

    for (int c = 0; c < 3; ++c) {
        float* cur = (c & 1) ? img1 : img0;
        float* nxt = (c & 1) ? img0 : img1;

        wait_async_done();          // channel c resident in `cur`
        __syncthreads();

        // Overlap: start DMA of channel c+1 into the other buffer, then compute.
        if (c + 1 < 3) issue_channel_load(xbase + (size_t)(c + 1) * 16384, nxt, t);

        // ---- Stage 1: R(128x16) = X(128x128) * F(128x16) via V_WMMA_F32_16X16X4_F32 ----
        // A 16x4 layout: lanes 0-15 hold K={k,k+1} in v0,v1; lanes 16-31 hold K={k+2,k+3}.
        // B 4x16 layout: v0 = rows {k (lanes 0-15), k+2 (lanes 16-31)}, v1 = {k+1, k+3}.
        v8f cacc = {};
        const int arow = w * 16 + lg;
        for (int k4 = 0; k4 < 128; k4 += 4) {
            const int ka = k4 + 2 * hi;
            v2f a;
            a.x = cur[arow * IMG_STRIDE + ka];
            a.y = cur[arow * IMG_STRIDE + ka + 1];
            v2f bb;
            bb.x = Fx[ka * 16 + lg];
            bb.y = Fx[(ka + 1) * 16 + lg];
            cacc = __builtin_amdgcn_wmma_f32_16x16x4_f32(
                false, a, false, bb, (short)0, cacc, false, false);
        }
        // D layout: VGPR r, lanes 0-15 -> M=r; lanes 16-31 -> M=r+8; N = lg.
#pragma unroll
        for (int r = 0; r < 8; ++r) {
            int row = w * 16 + r + 8 * hi;
            Rm[row * R_STRIDE + lg] = cacc[r];
        }
        __syncthreads();

        // ---- Stage 2: per-bin complex reduction over y (25 bins x 4 y-chunks) ----
        if (t < NBINS * 4) {
            int bin = t >> 2, pr = t & 3;
            int q2 = prm.binq[bin] * 2;
            int p2 = prm.binp[bin] * 2;
            float re = 0.0f, im = 0.0f;
            for (int i = 0; i < 32; ++i) {
                int y = pr * 32 + i;
                float rr = Rm[y * R_STRIDE + q2];
                float ri = Rm[y * R_STRIDE + q2 + 1];
                float cc = Gy[y * 16 + p2];
                float ss = Gy[y * 16 + p2 + 1];
                re += rr * cc - ri * ss;
                im += rr * ss + ri * cc;
            }
            part[bin * 8 + pr * 2]     = re;
            part[bin * 8 + pr * 2 + 1] = im;
        }
        __syncthreads();
        if (t < NBINS) {
            float re = part[t * 8 + 0] + part[t * 8 + 2] + part[t * 8 + 4] + part[t * 8 + 6];
            float im = part[t * 8 + 1] + part[t * 8 + 3] + part[t * 8 + 5] + part[t * 8 + 7];
            float mag = sqrtf(re * re + im * im);
            acc += log1pf(mag + 1e-12f);
        }
    }

    if (t < NBINS) out[(size_t)b * NBINS + t] = acc * (1.0f / 3.0f);
}

// ---------------- Host side ----------------

namespace {
struct BinItem { float r2; float ang; int idx; };

static void compute_params(FParams& p) {
    const int H = 128, W = 128;
    const float cy = (H - 1) * 0.5f, cx = (W - 1) * 0.5f;
    std::vector<BinItem> items(H * W);
    for (int y = 0; y < H; ++y) {
        for (int xx = 0; xx < W; ++xx) {
            float dy = (float)y - cy, dx = (float)xx - cx;
            BinItem& it = items[y * W + xx];
            it.r2  = dy * dy + dx * dx;
            it.ang = atan2f(dy, dx);
            it.idx = y * W + xx;
        }
    }
    // numpy: stable argsort by angle, then stable argsort by r2
    //  == lexicographic stable sort by (r2, ang) (ties in both are impossible).
    std::stable_sort(items.begin(), items.end(), [](const BinItem& a, const BinItem& b) {
        if (a.r2 != b.r2) return a.r2 < b.r2;
        return a.ang < b.ang;
    });

    p.Q = 0; p.P = 0;
    for (int i = 0; i < 8; ++i) { p.kxv[i] = 0; p.kyv[i] = 0; }
    for (int k = 0; k < NBINS; ++k) {
        int idx = items[k].idx;
        int sy = idx / W, sx = idx % W;
        int ky = (sy + H / 2) % H; if (ky >= H / 2) ky -= H;   // un-fftshift, signed freq
        int kx = (sx + W / 2) % W; if (kx >= W / 2) kx -= W;
        int qi = -1;
        for (int q = 0; q < p.Q; ++q) if (p.kxv[q] == kx) { qi = q; break; }
        if (qi < 0) { qi = p.Q; p.kxv[p.Q++] = kx; }
        int pi = -1;
        for (int q = 0; q < p.P; ++q) if (p.kyv[q] == ky) { pi = q; break; }
        if (pi < 0) { pi = p.P; p.kyv[p.P++] = ky; }
        p.binq[k] = qi;
        p.binp[k] = pi;
    }
}
} // namespace

extern "C" void kernel_launch(void* const* d_in, const int* in_sizes, int n_in,
                              void* d_out, int out_size, void* d_ws, size_t ws_size,
                              hipStream_t stream) {
    (void)n_in; (void)d_ws; (void)ws_size; (void)out_size;
    const float* x = (const float*)d_in[0];
    float* out = (float*)d_out;

    const int B = in_sizes[0] / (3 * 128 * 128);

    FParams prm;
    compute_params(prm);   // deterministic, recomputed every call (no caching)

    const size_t lds_floats = 2 * (128 * IMG_STRIDE) + 2048 + 2048 + 128 * R_STRIDE + 256;
    const size_t lds_bytes = lds_floats * sizeof(float);

    fftlow_kernel<<<dim3(B), dim3(256), lds_bytes, stream>>>(x, out, prm);
}